// GraphGRUForecasterMH_65377992179788
// MI455X (gfx1250) — compile-verified
//
#include <hip/hip_runtime.h>
#include <hip/hip_bf16.h>

// ---------------- problem constants (match reference) ----------------
#define Bsz    16
#define Wt     24
#define Nn     300
#define F_IN   6
#define H_HOR  24
#define F_NWP  10
#define Eg     3000
#define HG     64
#define HR     128
#define HD     128
#define NBIG   (Bsz*Wt*Nn)        // 115200 nodes in batched graph
#define ROWS   (Bsz*Nn)           // 4800 GRU rows
#define DROWS  (Bsz*H_HOR*Nn)     // 115200 decoder rows

typedef __attribute__((ext_vector_type(2))) float v2f;
typedef __attribute__((ext_vector_type(8))) float v8f;

__device__ __forceinline__ v8f wmma4(v2f a, v2f b, v8f c) {
  // D = A(16x4,f32) * B(4x16,f32) + C(16x16,f32), full fp32 — matches reference numerics
  return __builtin_amdgcn_wmma_f32_16x16x4_f32(false, a, false, b, (short)0, c, false, false);
}

// ---------------- graph template preprocessing (one 300-node graph) ----------------
__global__ void k_zero_counts(int* counts) {
  int i = threadIdx.x;
  if (i < Nn) counts[i] = 0;
}

__global__ void k_count(const long long* __restrict__ ei, int* __restrict__ counts, int E) {
  int e = blockIdx.x * blockDim.x + threadIdx.x;
  if (e < E) atomicAdd(&counts[(int)ei[E + e]], 1);   // col = target
}

__global__ void k_scan(const int* __restrict__ counts, int* __restrict__ offsets,
                       int* __restrict__ cursor, float* __restrict__ dinv) {
  if (threadIdx.x == 0) {
    int acc = 0;
    for (int i = 0; i < Nn; ++i) { offsets[i] = acc; cursor[i] = acc; acc += counts[i]; }
    offsets[Nn] = acc;
  }
  int i = threadIdx.x;
  if (i < Nn) dinv[i] = rsqrtf((float)(counts[i] + 1));  // +1 self loop
}

__global__ void k_fill(const long long* __restrict__ ei, int* __restrict__ cursor,
                       int* __restrict__ csr_src, int E) {
  int e = blockIdx.x * blockDim.x + threadIdx.x;
  if (e < E) {
    int c = (int)ei[E + e];
    int p = atomicAdd(&cursor[c], 1);
    csr_src[p] = (int)ei[e];
  }
}

// ---------------- GCN layer 1 feature transform: xw = X @ W1  (K=6, scalar) ----------------
__global__ void k_xw1(const float* __restrict__ X, const float* __restrict__ W1,
                      float* __restrict__ xw) {
  int idx = blockIdx.x * blockDim.x + threadIdx.x;
  if (idx >= NBIG * HG) return;
  int gn = idx >> 6, o = idx & 63;
  const float* x = X + (size_t)gn * F_IN;
  float s = 0.f;
#pragma unroll
  for (int k = 0; k < F_IN; ++k) s += x[k] * W1[k * HG + o];
  xw[idx] = s;
}

// ---------------- normalized scatter-gather aggregation (one wave per node) ----------------
__global__ void __launch_bounds__(256)
k_agg(const float* __restrict__ xw, const int* __restrict__ offsets,
      const int* __restrict__ csr_src, const float* __restrict__ dinv,
      const float* __restrict__ bias, float* __restrict__ out, int do_relu) {
  int gn   = (blockIdx.x * blockDim.x + threadIdx.x) >> 5;  // one wave32 per node
  int lane = threadIdx.x & 31;
  if (gn >= NBIG) return;
  int n = gn % Nn;
  const float* base = xw + (size_t)(gn - n) * HG;           // this graph copy's features
  float dn = dinv[n];
  float a0 = dn * dn * base[n * HG + lane];
  float a1 = dn * dn * base[n * HG + lane + 32];
  int s = offsets[n], e = offsets[n + 1];
  for (int i = s; i < e; ++i) {
    int src = csr_src[i];
    float w = dinv[src] * dn;
    a0 += w * base[src * HG + lane];
    a1 += w * base[src * HG + lane + 32];
  }
  a0 += bias[lane];      a1 += bias[lane + 32];
  if (do_relu) { a0 = fmaxf(a0, 0.f); a1 = fmaxf(a1, 0.f); }
  out[(size_t)gn * HG + lane]      = a0;
  out[(size_t)gn * HG + lane + 32] = a1;
}

// ---------------- GCN layer 2 GEMM: C[NBIG,64] = A[NBIG,64] @ W2[64,64] (WMMA fp32) ----------------
__global__ void __launch_bounds__(128)
k_gemm64(const float* __restrict__ A, const float* __restrict__ W2, float* __restrict__ C) {
  __shared__ float As[16 * 64];
  int tid = threadIdx.x, lane = tid & 31, wave = tid >> 5;
  int lhalf = lane >> 4, lidx = lane & 15;
  size_t row0 = (size_t)blockIdx.x * 16;
  for (int i = tid; i < 16 * 64; i += 128) As[i] = A[row0 * HG + i];
  __syncthreads();
  int jt = wave;                                  // 4 N-tiles of 16
  v8f c = {};
  for (int kt = 0; kt < 16; ++kt) {               // K = 64
    int k0 = kt * 4 + 2 * lhalf;
    v2f a; a.x = As[lidx * 64 + k0]; a.y = As[lidx * 64 + k0 + 1];
    v2f bv; bv.x = W2[(size_t)k0 * HG + jt * 16 + lidx];
            bv.y = W2[(size_t)(k0 + 1) * HG + jt * 16 + lidx];
    c = wmma4(a, bv, c);
  }
#pragma unroll
  for (int v = 0; v < 8; ++v) {
    int m = v + 8 * lhalf;
    C[(row0 + m) * HG + jt * 16 + lidx] = c[v];
  }
}

// ---------------- fused GRU: each block owns 16 rows, h lives in LDS for all 24 steps ----------------
__global__ void __launch_bounds__(256)
k_gru(const float* __restrict__ h2, const float* __restrict__ W_ih,
      const float* __restrict__ W_hh, const float* __restrict__ b_ih,
      const float* __restrict__ b_hh, float* __restrict__ hT) {
  __shared__ float hs[16 * HR];        // h state, 8 KB
  __shared__ float xs[16 * HG];        // x_t tile, 4 KB
  __shared__ float ghs[16 * 3 * HR];   // h-projection gates, 24 KB
  __shared__ float gxs[16 * 3 * HR];   // x-projection gates, 24 KB

  const int tid = threadIdx.x, lane = tid & 31, wave = tid >> 5;
  const int lhalf = lane >> 4, lidx = lane & 15;
  const int row0 = blockIdx.x * 16;    // rows r = b*300+n

  for (int i = tid; i < 16 * HR; i += 256) hs[i] = 0.f;
  __syncthreads();

  for (int t = 0; t < Wt; ++t) {
    // stage x_t: xs[m][k] = h2[((b*24 + t)*300 + n)*64 + k]
    for (int i = tid; i < 16 * HG; i += 256) {
      int m = i >> 6, k = i & 63;
      int r = row0 + m;
      int bb = r / Nn, n = r - bb * Nn;
      xs[i] = h2[(size_t)((bb * Wt + t) * Nn + n) * HG + k];
    }
    __syncthreads();

    // gates = h @ W_hh^T and x @ W_ih^T via WMMA; 24 N-tiles over 8 waves
    for (int jt = wave; jt < 24; jt += 8) {
      int j = jt * 16 + lidx;
      v8f ch = {}, cx = {};
      for (int kt = 0; kt < 32; ++kt) {          // K = 128 (hidden)
        int k0 = kt * 4 + 2 * lhalf;
        v2f a;  a.x = hs[lidx * HR + k0];        a.y = hs[lidx * HR + k0 + 1];
        v2f bv; bv.x = W_hh[(size_t)j * HR + k0]; bv.y = W_hh[(size_t)j * HR + k0 + 1];
        ch = wmma4(a, bv, ch);
      }
      for (int kt = 0; kt < 16; ++kt) {          // K = 64 (input)
        int k0 = kt * 4 + 2 * lhalf;
        v2f a;  a.x = xs[lidx * HG + k0];        a.y = xs[lidx * HG + k0 + 1];
        v2f bv; bv.x = W_ih[(size_t)j * HG + k0]; bv.y = W_ih[(size_t)j * HG + k0 + 1];
        cx = wmma4(a, bv, cx);
      }
#pragma unroll
      for (int v = 0; v < 8; ++v) {
        int m = v + 8 * lhalf;
        ghs[m * 384 + jt * 16 + lidx] = ch[v];
        gxs[m * 384 + jt * 16 + lidx] = cx[v];
      }
    }
    __syncthreads();

    // elementwise gate update (h_new depends only on same [m][j] of h_prev)
    for (int i = tid; i < 16 * HR; i += 256) {
      int m = i >> 7, j = i & 127;
      float xr = gxs[m * 384 + j]       + b_ih[j];
      float xz = gxs[m * 384 + 128 + j] + b_ih[128 + j];
      float xn = gxs[m * 384 + 256 + j] + b_ih[256 + j];
      float hr = ghs[m * 384 + j]       + b_hh[j];
      float hz = ghs[m * 384 + 128 + j] + b_hh[128 + j];
      float hn = ghs[m * 384 + 256 + j] + b_hh[256 + j];
      float rg = 1.f / (1.f + expf(-(xr + hr)));
      float zg = 1.f / (1.f + expf(-(xz + hz)));
      float ng = tanhf(xn + rg * hn);
      hs[i] = (1.f - zg) * ng + zg * hs[i];
    }
    __syncthreads();
  }

  for (int i = tid; i < 16 * HR; i += 256)
    hT[(size_t)(row0 + (i >> 7)) * HR + (i & 127)] = hs[i];
}

// ---------------- decoder: relu(dec_in @ Wd1^T + bd1) @ Wd2^T + bd2 (WMMA, K padded 138->140) --------
__global__ void __launch_bounds__(256)
k_dec(const float* __restrict__ hT, const float* __restrict__ Xf,
      const float* __restrict__ Wd1, const float* __restrict__ bd1,
      const float* __restrict__ Wd2, const float* __restrict__ bd2,
      float* __restrict__ out) {
  __shared__ float As[16 * 140];
  __shared__ float zs[16 * HD];
  const int tid = threadIdx.x, lane = tid & 31, wave = tid >> 5;
  const int lhalf = lane >> 4, lidx = lane & 15;
  const int row0 = blockIdx.x * 16;                 // rows ordered (b, h, n)

  for (int i = tid; i < 16 * 140; i += 256) {
    int m = i / 140, k = i - m * 140;
    int r = row0 + m;
    int bb = r / (H_HOR * Nn);
    int n  = r % Nn;
    float v;
    if (k < HR)            v = hT[(size_t)(bb * Nn + n) * HR + k];
    else if (k < HR + F_NWP) v = Xf[(size_t)r * F_NWP + (k - HR)];
    else                   v = 0.f;
    As[i] = v;
  }
  __syncthreads();

  int jt = wave;                                    // 8 N-tiles cover HD=128
  int j = jt * 16 + lidx;
  v8f c = {};
  for (int kt = 0; kt < 35; ++kt) {                 // K = 140 (padded)
    int k0 = kt * 4 + 2 * lhalf;
    v2f a;  a.x = As[lidx * 140 + k0]; a.y = As[lidx * 140 + k0 + 1];
    v2f bv;
    bv.x = (k0     < 138) ? Wd1[(size_t)j * 138 + k0]     : 0.f;
    bv.y = (k0 + 1 < 138) ? Wd1[(size_t)j * 138 + k0 + 1] : 0.f;
    c = wmma4(a, bv, c);
  }
#pragma unroll
  for (int v = 0; v < 8; ++v) {
    int m = v + 8 * lhalf;
    zs[m * HD + j] = fmaxf(c[v] + bd1[j], 0.f) * Wd2[j];
  }
  __syncthreads();
  if (tid < 16) {
    float s = 0.f;
    for (int k = 0; k < HD; ++k) s += zs[tid * HD + k];
    out[row0 + tid] = s + bd2[0];
  }
}

// ---------------- launcher ----------------
extern "C" void kernel_launch(void* const* d_in, const int* in_sizes, int n_in,
                              void* d_out, int out_size, void* d_ws, size_t ws_size,
                              hipStream_t stream) {
  const float*     X_seq = (const float*)d_in[0];
  const float*     X_fut = (const float*)d_in[1];
  const long long* eidx  = (const long long*)d_in[2];
  const float* W1   = (const float*)d_in[3];
  const float* b1   = (const float*)d_in[4];
  const float* W2   = (const float*)d_in[5];
  const float* b2   = (const float*)d_in[6];
  const float* W_ih = (const float*)d_in[7];
  const float* W_hh = (const float*)d_in[8];
  const float* b_ih = (const float*)d_in[9];
  const float* b_hh = (const float*)d_in[10];
  const float* Wd1  = (const float*)d_in[11];
  const float* bd1  = (const float*)d_in[12];
  const float* Wd2  = (const float*)d_in[13];
  const float* bd2  = (const float*)d_in[14];
  float* out = (float*)d_out;
  const int E = in_sizes[2] / 2;                    // 3000

  // workspace carve-up (~62 MB)
  float* wsf     = (float*)d_ws;
  float* xw_buf  = wsf;                             // NBIG*64
  float* h_buf   = xw_buf + (size_t)NBIG * HG;      // NBIG*64
  float* hT      = h_buf  + (size_t)NBIG * HG;      // ROWS*HR
  float* dinv    = hT     + (size_t)ROWS * HR;      // 300 (+pad)
  int*   counts  = (int*)(dinv + 320);
  int*   offsets = counts + Nn;                     // Nn+1
  int*   cursor  = offsets + (Nn + 4);
  int*   csr_src = cursor + Nn;                     // E

  // graph template preprocessing (rebuilt every call -> replay-deterministic)
  k_zero_counts<<<1, 320, 0, stream>>>(counts);
  k_count<<<(E + 255) / 256, 256, 0, stream>>>(eidx, counts, E);
  k_scan<<<1, 320, 0, stream>>>(counts, offsets, cursor, dinv);
  k_fill<<<(E + 255) / 256, 256, 0, stream>>>(eidx, cursor, csr_src, E);

  // GCN layer 1: transform + normalized aggregate + bias + relu
  k_xw1<<<(NBIG * HG + 255) / 256, 256, 0, stream>>>(X_seq, W1, xw_buf);
  k_agg<<<NBIG / 8, 256, 0, stream>>>(xw_buf, offsets, csr_src, dinv, b1, h_buf, 1);

  // GCN layer 2: WMMA GEMM + aggregate + bias
  k_gemm64<<<NBIG / 16, 128, 0, stream>>>(h_buf, W2, xw_buf);
  k_agg<<<NBIG / 8, 256, 0, stream>>>(xw_buf, offsets, csr_src, dinv, b2, h_buf, 0);

  // fused GRU over all 24 steps (WMMA gate GEMMs, h resident in LDS)
  k_gru<<<ROWS / 16, 256, 0, stream>>>(h_buf, W_ih, W_hh, b_ih, b_hh, hT);

  // decoder (WMMA) -> yhat [B, H, N, 1]
  k_dec<<<DROWS / 16, 256, 0, stream>>>(hT, X_fut, Wd1, bd1, Wd2, bd2, out);
}